// GIN_28183575396971
// MI455X (gfx1250) — compile-verified
//
#include <hip/hip_runtime.h>
#include <hip/hip_bf16.h>

typedef __attribute__((ext_vector_type(2))) float v2f;
typedef __attribute__((ext_vector_type(8))) float v8f;

#define DIM 128
#define LDSP 132          // padded LDS row stride (floats) to avoid bank conflicts
#define NGRAPH 64
#define BN_EPS 1e-5f

// ---------------------------------------------------------------- utilities
__global__ void zero_f32(float* __restrict__ p, int n) {
    int i = blockIdx.x * blockDim.x + threadIdx.x;
    if (i < n) p[i] = 0.0f;
}

__global__ void zero_i32(int* __restrict__ p, int n) {
    int i = blockIdx.x * blockDim.x + threadIdx.x;
    if (i < n) p[i] = 0;
}

// in-degree histogram (int), built once
__global__ void degi_kernel(const int* __restrict__ dst, int* __restrict__ degi, int E) {
    int e = blockIdx.x * blockDim.x + threadIdx.x;
    if (e < E) atomicAdd(&degi[dst[e]], 1);
}

// per-graph node counts, built once
__global__ void cnt_kernel(const int* __restrict__ batch, float* cnt, int N) {
    int i = blockIdx.x * blockDim.x + threadIdx.x;
    if (i < N) atomicAdd(&cnt[batch[i]], 1.0f);
}

// single-block exclusive scan of degi -> offs[0..N], offs[N] = E
__global__ void scan_kernel(const int* __restrict__ degi, int* __restrict__ offs, int N) {
    __shared__ int buf[256];
    __shared__ int carry;
    const int tid = threadIdx.x;
    if (tid == 0) carry = 0;
    __syncthreads();
    for (int base = 0; base < N; base += 256) {
        int i = base + tid;
        int v = (i < N) ? degi[i] : 0;
        buf[tid] = v;
        __syncthreads();
        #pragma unroll
        for (int off = 1; off < 256; off <<= 1) {
            int t = (tid >= off) ? buf[tid - off] : 0;
            __syncthreads();
            buf[tid] += t;
            __syncthreads();
        }
        int incl = buf[tid];
        if (i < N) offs[i] = carry + incl - v;   // exclusive
        __syncthreads();
        if (tid == 0) carry += buf[255];
        __syncthreads();
    }
    if (tid == 0) offs[N] = carry;
}

__global__ void copy_i32(const int* __restrict__ src, int* __restrict__ dst, int n) {
    int i = blockIdx.x * blockDim.x + threadIdx.x;
    if (i < n) dst[i] = src[i];
}

// bucket edges by dst: csr_src[offs[dst] .. offs[dst]+deg) = src ids
__global__ void fill_csr(const int* __restrict__ src, const int* __restrict__ dst,
                         int* cursor, int* __restrict__ csr, int E) {
    int e = blockIdx.x * blockDim.x + threadIdx.x;
    if (e < E) {
        int p = atomicAdd(&cursor[dst[e]], 1);
        csr[p] = src[e];
    }
}

// One wave per node: agg[node] = sum over in-neighbors of h[src].
// Lane holds a float4 slice (32 lanes * 4 = 128). Pure gather, no atomics.
__global__ void __launch_bounds__(256) gather_kernel(
    const float* __restrict__ h, const int* __restrict__ offs,
    const int* __restrict__ csr, float* __restrict__ agg, int N)
{
    int gid  = blockIdx.x * 256 + threadIdx.x;
    int node = gid >> 5;
    int lane = threadIdx.x & 31;
    if (node >= N) return;
    int beg = offs[node], end = offs[node + 1];
    float4 acc = make_float4(0.f, 0.f, 0.f, 0.f);
    for (int i = beg; i < end; ++i) {
        int s = csr[i];
        const float4 v = ((const float4*)(h + (size_t)s * DIM))[lane];
        acc.x += v.x; acc.y += v.y; acc.z += v.z; acc.w += v.w;
    }
    ((float4*)(agg + (size_t)node * DIM))[lane] = acc;
}

// ---------------------------------------------------------------- GEMM 1
// z = h + agg/deg ;  z1 = z @ W + b ; accumulate per-column sum / sumsq for BN.
__global__ void __launch_bounds__(256) gemm1_kernel(
    const float* __restrict__ hin, const float* agg,
    const int* __restrict__ degi, const float* __restrict__ W,
    const float* __restrict__ bias, float* z1,
    float* __restrict__ colsum, float* __restrict__ colsq)
{
    __shared__ float zt[16 * LDSP];
    const int row0 = blockIdx.x * 16;
    const int tid  = threadIdx.x;

    for (int i = tid; i < 16 * DIM; i += 256) {
        int r = i >> 7, c = i & (DIM - 1);
        int gr = row0 + r;
        float dg = fmaxf((float)degi[gr], 1.0f);
        zt[r * LDSP + c] = hin[(size_t)gr * DIM + c] + agg[(size_t)gr * DIM + c] / dg;
    }
    __syncthreads();

    const int wave = tid >> 5, lane = tid & 31;
    const int m  = lane & 15;            // A row / B col within tile
    const int kb = (lane >> 4) << 1;     // 0 or 2 (K sub-slot)
    const int n  = wave * 16 + m;        // global output column

    v8f acc = {};
    #pragma unroll 8
    for (int kk = 0; kk < DIM; kk += 4) {
        v2f a, b;
        a.x = zt[m * LDSP + kk + kb];
        a.y = zt[m * LDSP + kk + kb + 1];
        b.x = W[(kk + kb)     * DIM + n];
        b.y = W[(kk + kb + 1) * DIM + n];
        acc = __builtin_amdgcn_wmma_f32_16x16x4_f32(
            false, a, false, b, (short)0, acc, false, false);
    }

    const float bs = bias[n];
    const int rbase = (lane >> 4) << 3;  // rows 0..7 or 8..15
    float s = 0.f, sq = 0.f;
    #pragma unroll
    for (int v = 0; v < 8; ++v) {
        float val = acc[v] + bs;
        z1[(size_t)(row0 + rbase + v) * DIM + n] = val;
        s += val; sq += val * val;
    }
    s  += __shfl_xor(s,  16, 32);        // combine rows 0-7 with 8-15 (same column)
    sq += __shfl_xor(sq, 16, 32);
    if (lane < 16) {
        atomicAdd(&colsum[n], s);
        atomicAdd(&colsq[n],  sq);
    }
}

// ---------------------------------------------------------------- BN finalize
__global__ void bn_finalize(const float* __restrict__ gamma, const float* __restrict__ beta,
                            float* colsum, float* colsq,
                            float* __restrict__ bnsc, float* __restrict__ bnsh, int N)
{
    int c = threadIdx.x;
    if (c < DIM) {
        float inv = 1.0f / (float)N;
        float mu   = colsum[c] * inv;
        float var  = colsq[c] * inv - mu * mu;
        float rstd = rsqrtf(var + BN_EPS);
        float g = gamma[c];
        bnsc[c] = g * rstd;
        bnsh[c] = beta[c] - g * mu * rstd;
        colsum[c] = 0.0f;  // re-arm for next layer
        colsq[c]  = 0.0f;
    }
}

// ---------------------------------------------------------------- GEMM 2
// y = relu(bnsc*z1 + bnsh) ; h = y @ W2 + b2 ; node_pool += h ; gsum[batch] += h.
__global__ void __launch_bounds__(256) gemm2_kernel(
    const float* z1, const float* __restrict__ bnsc, const float* __restrict__ bnsh,
    const float* __restrict__ W, const float* __restrict__ bias,
    const int* __restrict__ batch,
    float* __restrict__ hout, float* __restrict__ node_pool, float* __restrict__ gsum)
{
    __shared__ float zt[16 * LDSP];
    const int row0 = blockIdx.x * 16;
    const int tid  = threadIdx.x;

    for (int i = tid; i < 16 * DIM; i += 256) {
        int r = i >> 7, c = i & (DIM - 1);
        float val = z1[(size_t)(row0 + r) * DIM + c] * bnsc[c] + bnsh[c];
        zt[r * LDSP + c] = fmaxf(val, 0.0f);
    }
    __syncthreads();

    const int wave = tid >> 5, lane = tid & 31;
    const int m  = lane & 15;
    const int kb = (lane >> 4) << 1;
    const int n  = wave * 16 + m;

    v8f acc = {};
    #pragma unroll 8
    for (int kk = 0; kk < DIM; kk += 4) {
        v2f a, b;
        a.x = zt[m * LDSP + kk + kb];
        a.y = zt[m * LDSP + kk + kb + 1];
        b.x = W[(kk + kb)     * DIM + n];
        b.y = W[(kk + kb + 1) * DIM + n];
        acc = __builtin_amdgcn_wmma_f32_16x16x4_f32(
            false, a, false, b, (short)0, acc, false, false);
    }

    const float bs = bias[n];
    const int rbase = (lane >> 4) << 3;
    const int b0 = batch[row0];
    const bool uniform = (batch[row0 + 15] == b0);  // batch ids are sorted

    float s = 0.f;
    #pragma unroll
    for (int v = 0; v < 8; ++v) {
        int gr = row0 + rbase + v;
        float val = acc[v] + bs;
        size_t idx = (size_t)gr * DIM + n;
        hout[idx] = val;
        node_pool[idx] += val;
        if (uniform) s += val;
        else atomicAdd(&gsum[(size_t)batch[gr] * DIM + n], val);
    }
    if (uniform) {
        s += __shfl_xor(s, 16, 32);
        if (lane < 16) atomicAdd(&gsum[(size_t)b0 * DIM + n], s);
    }
}

__global__ void finalize_gpool(const float* __restrict__ gsum,
                               const float* __restrict__ cnt, float* __restrict__ out)
{
    int i = blockIdx.x * blockDim.x + threadIdx.x;
    if (i < NGRAPH * DIM) {
        int g = i >> 7;
        out[i] = gsum[i] / fmaxf(cnt[g], 1.0f);
    }
}

// ---------------------------------------------------------------- launch
extern "C" void kernel_launch(void* const* d_in, const int* in_sizes, int n_in,
                              void* d_out, int out_size, void* d_ws, size_t ws_size,
                              hipStream_t stream)
{
    const float* x     = (const float*)d_in[0];
    const int*   edge  = (const int*)  d_in[1];
    const int*   batch = (const int*)  d_in[2];
    const float* W1    = (const float*)d_in[3];
    const float* b1    = (const float*)d_in[4];
    const float* gamma = (const float*)d_in[5];
    const float* beta  = (const float*)d_in[6];
    const float* W2    = (const float*)d_in[7];
    const float* b2    = (const float*)d_in[8];

    const int N = in_sizes[0] / DIM;              // 50000 (divisible by 16)
    const int E = in_sizes[1] / 2;                // 800000
    const int L = in_sizes[3] / (DIM * DIM);      // 4
    const int* src = edge;
    const int* dst = edge + E;

    // ---- workspace layout (floats then ints) ----
    float* ws    = (float*)d_ws;
    float* agg   = ws;                                 // N*DIM (aliased as z1)
    float* hbuf  = ws + (size_t)N * DIM;               // N*DIM
    float* fsm   = hbuf + (size_t)N * DIM;             // small float state
    float* cnt    = fsm;                               // NGRAPH
    float* colsum = cnt + NGRAPH;                      // DIM
    float* colsq  = colsum + DIM;                      // DIM
    float* bnsc   = colsq + DIM;                       // DIM
    float* bnsh   = bnsc + DIM;                        // DIM
    float* gsum   = bnsh + DIM;                        // NGRAPH*DIM
    int*   ints   = (int*)(gsum + NGRAPH * DIM);
    int*   degi   = ints;                              // N
    int*   offs   = degi + N;                          // N+1
    int*   cursor = offs + N + 1;                      // N
    int*   csr    = cursor + N;                        // E

    float* node_pool = (float*)d_out;                  // N*DIM
    float* gpool_out = node_pool + (size_t)N * DIM;    // NGRAPH*DIM

    const int ND = N * DIM;
    const int fsmN = NGRAPH + 4 * DIM + NGRAPH * DIM;

    // ---- one-time state: zeros, degrees, graph counts, CSR ----
    zero_f32<<<(fsmN + 255) / 256, 256, 0, stream>>>(fsm, fsmN);
    zero_f32<<<(ND + 255) / 256, 256, 0, stream>>>(node_pool, ND);
    zero_i32<<<(N + 255) / 256, 256, 0, stream>>>(degi, N);
    degi_kernel<<<(E + 255) / 256, 256, 0, stream>>>(dst, degi, E);
    cnt_kernel<<<(N + 255) / 256, 256, 0, stream>>>(batch, cnt, N);
    scan_kernel<<<1, 256, 0, stream>>>(degi, offs, N);
    copy_i32<<<(N + 255) / 256, 256, 0, stream>>>(offs, cursor, N);
    fill_csr<<<(E + 255) / 256, 256, 0, stream>>>(src, dst, cursor, csr, E);

    // ---- layers ----
    const int tiles = N / 16;
    for (int l = 0; l < L; ++l) {
        const float* hin = (l == 0) ? x : hbuf;
        gather_kernel<<<(N * 32 + 255) / 256, 256, 0, stream>>>(hin, offs, csr, agg, N);
        gemm1_kernel<<<tiles, 256, 0, stream>>>(
            hin, agg, degi, W1 + (size_t)l * DIM * DIM, b1 + l * DIM,
            agg /* z1 aliases agg */, colsum, colsq);
        bn_finalize<<<1, DIM, 0, stream>>>(gamma + l * DIM, beta + l * DIM,
                                           colsum, colsq, bnsc, bnsh, N);
        gemm2_kernel<<<tiles, 256, 0, stream>>>(
            agg /* z1 */, bnsc, bnsh, W2 + (size_t)l * DIM * DIM, b2 + l * DIM,
            batch, hbuf, node_pool, gsum);
    }
    finalize_gpool<<<(NGRAPH * DIM + 255) / 256, 256, 0, stream>>>(gsum, cnt, gpool_out);
}